// GraphCritic_d_46454366273713
// MI455X (gfx1250) — compile-verified
//
#include <hip/hip_runtime.h>

typedef __attribute__((ext_vector_type(2))) float v2f;
typedef __attribute__((ext_vector_type(8))) float v8f;
typedef int v4i_ __attribute__((vector_size(16)));  // matches builtin's V4i

#if __has_builtin(__builtin_amdgcn_global_load_async_to_lds_b128)
#define USE_ASYNC_STAGE 1
#else
#define USE_ASYNC_STAGE 0
#endif

// ---------------------------------------------------------------------------
// Small per-node-type MLP:  out[r, 0:16] = tanh(x[r, 0:IN] @ W + b)
// One thread per row; W/b are uniform -> scalar loads; 16-float row stores.
// ---------------------------------------------------------------------------
template <int IN>
__global__ __launch_bounds__(256) void node_mlp_kernel(
    const float* __restrict__ x, const float* __restrict__ W,
    const float* __restrict__ b, float* __restrict__ out, int n) {
  int r = blockIdx.x * blockDim.x + threadIdx.x;
  if (r >= n) return;
  float xi[IN];
#pragma unroll
  for (int k = 0; k < IN; ++k) xi[k] = x[(size_t)r * IN + k];
#pragma unroll
  for (int c = 0; c < 16; ++c) {
    float acc = b[c];
#pragma unroll
    for (int k = 0; k < IN; ++k) acc += xi[k] * W[k * 16 + c];
    out[(size_t)r * 16 + c] = tanhf(acc);
  }
}

// ---------------------------------------------------------------------------
// Edge scatter: pas_sum[send[e]] += pas_feat[recv[e]]; pas_cnt[send[e]] += 1
// One thread per edge; 128-bit gathers (pas_feat rows are 64B aligned),
// f32 global atomics for the scatter (global_atomic_add_f32).
// ---------------------------------------------------------------------------
__global__ __launch_bounds__(256) void edge_scatter_kernel(
    const float* __restrict__ pas_feat, const int* __restrict__ recv,
    const int* __restrict__ send, float* __restrict__ pas_sum,
    float* __restrict__ pas_cnt, int E) {
  int e = blockIdx.x * blockDim.x + threadIdx.x;
  if (e >= E) return;
  int p = recv[e];
  int v = send[e];
  const float4* src = (const float4*)(pas_feat + (size_t)p * 16);
  float* dst = pas_sum + (size_t)v * 16;
#pragma unroll
  for (int i = 0; i < 4; ++i) {
    float4 f = src[i];
    atomicAdd(dst + i * 4 + 0, f.x);
    atomicAdd(dst + i * 4 + 1, f.y);
    atomicAdd(dst + i * 4 + 2, f.z);
    atomicAdd(dst + i * 4 + 3, f.w);
  }
  atomicAdd(pas_cnt + v, 1.0f);
}

__device__ __forceinline__ int lower_bound_i(const int* __restrict__ a, int n, int v) {
  int lo = 0, hi = n;
  while (lo < hi) {
    int mid = (lo + hi) >> 1;
    if (a[mid] < v) lo = mid + 1; else hi = mid;
  }
  return lo;
}

// ---------------------------------------------------------------------------
// Sorted-segment mean with fused tanh-MLP (requests path).
// One wave32 per batch element: binary-search the segment, lanes 0..15 own
// channels of even rows, lanes 16..31 own channels of odd rows, combine via
// __shfl_xor(.,16). Deterministic (no atomics).
// ---------------------------------------------------------------------------
template <int IN>
__global__ __launch_bounds__(256) void seg_mean_mlp_kernel(
    const float* __restrict__ x, const int* __restrict__ batch, int nrows,
    const float* __restrict__ W, const float* __restrict__ b,
    float* __restrict__ out_mean, int Bn) {
  int wave = (blockIdx.x * blockDim.x + threadIdx.x) >> 5;
  int lane = threadIdx.x & 31;
  if (wave >= Bn) return;
  int start = lower_bound_i(batch, nrows, wave);
  int end = lower_bound_i(batch, nrows, wave + 1);
  int c = lane & 15, hf = lane >> 4;
  float wcol[IN];
#pragma unroll
  for (int k = 0; k < IN; ++k) wcol[k] = W[k * 16 + c];
  float bias = b[c];
  float s = 0.0f;
  for (int r = start + hf; r < end; r += 2) {
    float a = bias;
#pragma unroll
    for (int k = 0; k < IN; ++k) a += x[(size_t)r * IN + k] * wcol[k];
    s += tanhf(a);
  }
  s += __shfl_xor(s, 16, 32);
  float cnt = (float)(end - start);
  if (lane < 16) out_mean[(size_t)wave * 16 + c] = s / fmaxf(cnt, 1.0f);
}

// Sorted-segment mean over a materialized [n,16] feature (vehicles path).
__global__ __launch_bounds__(256) void seg_mean_feat_kernel(
    const float* __restrict__ feat, const int* __restrict__ batch, int nrows,
    float* __restrict__ out_mean, int Bn) {
  int wave = (blockIdx.x * blockDim.x + threadIdx.x) >> 5;
  int lane = threadIdx.x & 31;
  if (wave >= Bn) return;
  int start = lower_bound_i(batch, nrows, wave);
  int end = lower_bound_i(batch, nrows, wave + 1);
  int c = lane & 15, hf = lane >> 4;
  float s = 0.0f;
  for (int r = start + hf; r < end; r += 2) s += feat[(size_t)r * 16 + c];
  s += __shfl_xor(s, 16, 32);
  float cnt = (float)(end - start);
  if (lane < 16) out_mean[(size_t)wave * 16 + c] = s / fmaxf(cnt, 1.0f);
}

// ---------------------------------------------------------------------------
// Assemble action_feat[b, 0:64] = [req_mean | veh_feat[idx] | pas_mean[idx] | veh_mean]
// ---------------------------------------------------------------------------
__global__ __launch_bounds__(256) void assemble_kernel(
    const float* __restrict__ req_mean, const float* __restrict__ veh_feat,
    const float* __restrict__ pas_sum, const float* __restrict__ pas_cnt,
    const float* __restrict__ veh_mean, const int* __restrict__ r2v,
    float* __restrict__ action, int Bn) {
  int bi = blockIdx.x * blockDim.x + threadIdx.x;
  if (bi >= Bn) return;
  int idx = r2v[bi];
  float inv = 1.0f / fmaxf(pas_cnt[idx], 1.0f);
  float4* o4 = (float4*)(action + (size_t)bi * 64);
  const float4* rm = (const float4*)(req_mean + (size_t)bi * 16);
  const float4* vf = (const float4*)(veh_feat + (size_t)idx * 16);
  const float4* ps = (const float4*)(pas_sum + (size_t)idx * 16);
  const float4* vm = (const float4*)(veh_mean + (size_t)bi * 16);
#pragma unroll
  for (int i = 0; i < 4; ++i) o4[i] = rm[i];
#pragma unroll
  for (int i = 0; i < 4; ++i) o4[4 + i] = vf[i];
#pragma unroll
  for (int i = 0; i < 4; ++i) {
    float4 t = ps[i];
    t.x *= inv; t.y *= inv; t.z *= inv; t.w *= inv;
    o4[8 + i] = t;
  }
#pragma unroll
  for (int i = 0; i < 4; ++i) o4[12 + i] = vm[i];
}

// ---------------------------------------------------------------------------
// Pack W1/W2 (64x64 row-major) into WMMA B-fragment order so the GEMM loop
// does ONE coalesced 64-bit load per v_wmma:
//   Wp[layer*2048 + nb*512 + kb*32 + lane] = { W[k][n], W[k+1][n] }
//   with m = lane%16, hi = lane/16, k = kb*4 + hi*2, n = nb*16 + m.
// ---------------------------------------------------------------------------
__global__ __launch_bounds__(256) void pack_w_kernel(
    const float* __restrict__ W1, const float* __restrict__ W2,
    float2* __restrict__ Wp) {
  int t = blockIdx.x * blockDim.x + threadIdx.x;
  if (t >= 4096) return;
  int lane = t & 31;
  int kb = (t >> 5) & 15;
  int nb = (t >> 9) & 3;
  int layer = (t >> 11) & 1;
  const float* W = layer ? W2 : W1;
  int m = lane & 15, hi = lane >> 4;
  int k = kb * 4 + hi * 2;
  int n = nb * 16 + m;
  Wp[t] = make_float2(W[k * 64 + n], W[(k + 1) * 64 + n]);
}

// ---------------------------------------------------------------------------
// MLP head with f32 WMMA (V_WMMA_F32_16X16X4_F32).
// One wave32 per 16-row tile; 8 waves / block. Activations staged in LDS with
// a 68-float row pitch (272B = 17*16B: 128-bit aligned, and rotates banks by
// 4 per row so column-strided A-fragment reads hit 16 distinct banks).
// Staging uses GLOBAL_LOAD_ASYNC_TO_LDS_B128 when the toolchain exposes it.
//   A frag (16x4 f32): lane l -> row M=l%16, k-offset (l/16)*2, 2 VGPRs.
//   B frag (4x16 f32): packed float2 per lane (see pack_w_kernel).
//   D (16x16 f32):     v8f, reg r -> row r + 8*(l/16), col l%16.
// ---------------------------------------------------------------------------
__global__ __launch_bounds__(256) void mlp_head_kernel(
    const float* __restrict__ action, const float2* __restrict__ Wp,
    const float* __restrict__ b1, const float* __restrict__ b2,
    const float* __restrict__ W3, const float* __restrict__ b3,
    float* __restrict__ out, int Bn) {
  __shared__ float lds[8][16 * 68];
  const int wave = threadIdx.x >> 5;
  const int lane = threadIdx.x & 31;
  float* tile = &lds[wave][0];
  const int tiles = Bn >> 4;
  int tileIdx = blockIdx.x * 8 + wave;
  if (tileIdx >= tiles) tileIdx = tiles - 1;  // keep waves converged for syncs
  const int row0 = tileIdx * 16;

  // Warm the packed weights into cache (global_prefetch_b8).
  __builtin_prefetch((const char*)Wp + (threadIdx.x & 255) * 64, 0, 0);

  // Stage 16x64 activation tile -> LDS.
#if USE_ASYNC_STAGE
#pragma unroll
  for (int i = 0; i < 8; ++i) {
    int flat = i * 32 + lane;            // float4 chunk id, 0..255
    int r = flat >> 4, c4 = flat & 15;   // row, float4 column
    const float* gsrc = action + (size_t)(row0 + r) * 64 + c4 * 4;
    __builtin_amdgcn_global_load_async_to_lds_b128(
        (__attribute__((address_space(1))) v4i_*)gsrc,
        (__attribute__((address_space(3))) v4i_*)(tile + r * 68 + c4 * 4),
        0, 0);
  }
#if __has_builtin(__builtin_amdgcn_s_wait_asynccnt)
  __builtin_amdgcn_s_wait_asynccnt(0);
#else
  asm volatile("s_wait_asynccnt 0x0" ::: "memory");
#endif
#else
#pragma unroll
  for (int i = 0; i < 32; ++i) {
    int flat = i * 32 + lane;
    int r = flat >> 6, c = flat & 63;
    tile[r * 68 + c] = action[(size_t)(row0 + r) * 64 + c];
  }
#endif
  __syncthreads();

  const int m = lane & 15;
  const int hi = lane >> 4;
  const float* bs[2] = {b1, b2};

  for (int layer = 0; layer < 2; ++layer) {
    const float2* __restrict__ Wl = Wp + (size_t)layer * 2048;
    const float* __restrict__ bb = bs[layer];
    v8f acc[4];
#pragma unroll
    for (int nb = 0; nb < 4; ++nb) {
      v8f c = {};
#pragma unroll
      for (int kb = 0; kb < 16; ++kb) {
        const int k = kb * 4 + hi * 2;
        v2f a;
        a.x = tile[m * 68 + k];
        a.y = tile[m * 68 + k + 1];
        float2 w = Wl[nb * 512 + kb * 32 + lane];
        v2f bf;
        bf.x = w.x;
        bf.y = w.y;
        c = __builtin_amdgcn_wmma_f32_16x16x4_f32(
            /*neg_a=*/false, a, /*neg_b=*/false, bf,
            /*c_mod=*/(short)0, c, /*reuse_a=*/false, /*reuse_b=*/false);
      }
      acc[nb] = c;
    }
    __syncthreads();  // all A-fragment reads done before LDS is overwritten
#pragma unroll
    for (int nb = 0; nb < 4; ++nb) {
      const int n = nb * 16 + m;
      const float bv = bb[n];
#pragma unroll
      for (int r = 0; r < 8; ++r) {
        const int rr = r + 8 * hi;
        tile[rr * 68 + n] = tanhf(acc[nb][r] + bv);
      }
    }
    __syncthreads();
  }

  // Final 64 -> 1: lanes 0..15 each reduce one row.
  if (lane < 16) {
    float s = b3[0];
#pragma unroll
    for (int c = 0; c < 64; ++c) s += tile[lane * 68 + c] * W3[c];
    out[row0 + lane] = s;
  }
}

// ---------------------------------------------------------------------------
// Host-side launcher
// ---------------------------------------------------------------------------
extern "C" void kernel_launch(void* const* d_in, const int* in_sizes, int n_in,
                              void* d_out, int out_size, void* d_ws, size_t ws_size,
                              hipStream_t stream) {
  const float* requests_x = (const float*)d_in[0];
  const int* requests_b = (const int*)d_in[1];
  const float* vehicles_x = (const float*)d_in[2];
  const int* vehicles_b = (const int*)d_in[3];
  const float* passengers_x = (const float*)d_in[4];
  const int* e_recv = (const int*)d_in[5];
  const int* e_send = (const int*)d_in[6];
  const int* r2v = (const int*)d_in[7];
  const float* W_req = (const float*)d_in[8];
  const float* b_req = (const float*)d_in[9];
  const float* W_veh = (const float*)d_in[10];
  const float* b_veh = (const float*)d_in[11];
  const float* W_pas = (const float*)d_in[12];
  const float* b_pas = (const float*)d_in[13];
  const float* W1 = (const float*)d_in[14];
  const float* b1 = (const float*)d_in[15];
  const float* W2 = (const float*)d_in[16];
  const float* b2 = (const float*)d_in[17];
  const float* W3 = (const float*)d_in[18];
  const float* b3 = (const float*)d_in[19];

  const int NREQ = in_sizes[1];
  const int NVEH = in_sizes[3];
  const int NPAS = in_sizes[4] / 10;
  const int E = in_sizes[5];
  const int Bn = in_sizes[7];

  // Workspace carve-out (floats, 256B-aligned chunks).
  float* ws = (float*)d_ws;
  size_t off = 0;
  auto carve = [&](size_t nflt) {
    float* p = ws + off;
    off += (nflt + 63) & ~(size_t)63;
    return p;
  };
  float* pasFeat = carve((size_t)NPAS * 16);
  float* vehFeat = carve((size_t)NVEH * 16);
  float* pasSum = carve((size_t)NVEH * 16);
  float* pasCnt = carve((size_t)NVEH);
  float* reqMean = carve((size_t)Bn * 16);
  float* vehMean = carve((size_t)Bn * 16);
  float* action = carve((size_t)Bn * 64);
  float* wPack = carve(8192);  // 2 layers * 4 nb * 16 kb * 32 lanes * float2
  if (off * sizeof(float) > ws_size) return;  // insufficient scratch

  // Zero the scatter accumulators.
  (void)hipMemsetAsync(pasSum, 0, (size_t)NVEH * 16 * sizeof(float), stream);
  (void)hipMemsetAsync(pasCnt, 0, (size_t)NVEH * sizeof(float), stream);

  // 0) Pack head weights into WMMA fragment order.
  pack_w_kernel<<<16, 256, 0, stream>>>(W1, W2, (float2*)wPack);

  // 1) Node-type MLPs.
  node_mlp_kernel<10><<<(NPAS + 255) / 256, 256, 0, stream>>>(
      passengers_x, W_pas, b_pas, pasFeat, NPAS);
  node_mlp_kernel<8><<<(NVEH + 255) / 256, 256, 0, stream>>>(
      vehicles_x, W_veh, b_veh, vehFeat, NVEH);

  // 2) Edge gather + scatter-add onto vehicles.
  edge_scatter_kernel<<<(E + 255) / 256, 256, 0, stream>>>(
      pasFeat, e_recv, e_send, pasSum, pasCnt, E);

  // 3) Sorted-segment means (wave32 per batch element; 8 waves/block).
  seg_mean_mlp_kernel<10><<<(Bn * 32 + 255) / 256, 256, 0, stream>>>(
      requests_x, requests_b, NREQ, W_req, b_req, reqMean, Bn);
  seg_mean_feat_kernel<<<(Bn * 32 + 255) / 256, 256, 0, stream>>>(
      vehFeat, vehicles_b, NVEH, vehMean, Bn);

  // 4) Assemble [B,64] action features.
  assemble_kernel<<<(Bn + 255) / 256, 256, 0, stream>>>(
      reqMean, vehFeat, pasSum, pasCnt, vehMean, r2v, action, Bn);

  // 5) WMMA MLP head -> d_out [B].
  const int tiles = Bn / 16;
  const int blocks = (tiles + 7) / 8;
  mlp_head_kernel<<<blocks, 256, 0, stream>>>(
      action, (const float2*)wPack, b1, b2, W3, b3, (float*)d_out, Bn);
}